// slater_jastrow_25735444038448
// MI455X (gfx1250) — compile-verified
//
#include <hip/hip_runtime.h>
#include <hip/hip_bf16.h>
#include <math.h>

#define B_TOT 16384
#define N_ORB 64
#define NF    32
#define NSPIN 2
#define NV    128   // NSPIN * N_ORB

typedef __attribute__((ext_vector_type(2))) float v2f;
typedef __attribute__((ext_vector_type(8))) float v8f;

// ---------------------------------------------------------------------------
// Kernel A: G[i][j] = 4 * Re(W[i][j]).  W is complex64 interleaved.
// y_b = n^T (2*(ReW + ReW^T)) n == 4 * n^T ReW n, so fold the 4 into G.
// ---------------------------------------------------------------------------
__global__ void prep_g(const float* __restrict__ Wf, float* __restrict__ G) {
    int i = blockIdx.x * blockDim.x + threadIdx.x;
    if (i < NV * NV) G[i] = 4.0f * Wf[2 * i];
}

// ---------------------------------------------------------------------------
// Kernel B: y_b = sum_j (n @ G)[b][j] * n[b][j], via V_WMMA_F32_16X16X4_F32.
// One wave handles 16 batch rows. For each 16-col tile of G we chain 32
// k-step WMMAs (K=4 each), then multiply by the n tile and reduce.
//
// fp32 A-layout (16x4): lanes 0-15 -> M=lane, regs {K+0,K+1};
//                       lanes 16-31 -> M=lane-16, regs {K+2,K+3}.
// fp32 B-layout (4x16): mirrored (lanes 0-15: rows K+0/K+1 at N=lane;
//                       lanes 16-31: rows K+2/K+3).
// fp32 C/D (16x16): reg r, lanes 0-15 -> row r, lanes 16-31 -> row r+8.
// ---------------------------------------------------------------------------
__global__ __launch_bounds__(32) void quad_wmma(const float* __restrict__ n,
                                                const float* __restrict__ G,
                                                float* __restrict__ yout) {
    const int lane = threadIdx.x;
    const int half = lane >> 4;      // 0: K pair {0,1}, 1: K pair {2,3}
    const int l    = lane & 15;
    const int b0   = blockIdx.x * 16;

    __builtin_prefetch(G + lane * 4, 0, 0);   // global_prefetch of the hot matrix

    float ysum[8];
#pragma unroll
    for (int r = 0; r < 8; ++r) ysum[r] = 0.f;

    for (int ct = 0; ct < 8; ++ct) {
        v8f acc = {};
#pragma unroll 4
        for (int kk = 0; kk < NV; kk += 4) {
            // A operand: n[b0+l][kk + 2*half + {0,1}]
            const float* ap = n + (size_t)(b0 + l) * NV + kk + 2 * half;
            v2f a; a.x = ap[0]; a.y = ap[1];
            // B operand: G[kk + 2*half + {0,1}][ct*16 + l]
            const float* bp = G + (size_t)(kk + 2 * half) * NV + ct * 16 + l;
            v2f bv; bv.x = bp[0]; bv.y = bp[NV];
            acc = __builtin_amdgcn_wmma_f32_16x16x4_f32(
                false, a, false, bv, (short)0, acc, false, false);
        }
        // acc[r] = t[row = r + 8*half][col = ct*16 + l]; dot against n.
#pragma unroll
        for (int r = 0; r < 8; ++r) {
            float nv = n[(size_t)(b0 + r + 8 * half) * NV + ct * 16 + l];
            ysum[r] += acc[r] * nv;
        }
    }
    // Reduce across the 16 lanes of each half (xor masks 1..8 stay in-half).
#pragma unroll
    for (int r = 0; r < 8; ++r) {
        float v = ysum[r];
        for (int m = 1; m <= 8; m <<= 1) v += __shfl_xor(v, m, 32);
        ysum[r] = v;
    }
    if (l == 0) {
#pragma unroll
        for (int r = 0; r < 8; ++r) yout[b0 + r + 8 * half] = ysum[r];
    }
}

// ---------------------------------------------------------------------------
// Kernel C: per batch: two 32x32 complex log-dets (one per spin) via LU with
// partial pivoting in LDS, one wave per spin; combine with y, emit complex64.
// ---------------------------------------------------------------------------
__device__ inline float2 cdivf2(float2 a, float2 b) {
    float inv = 1.0f / (b.x * b.x + b.y * b.y);
    return make_float2((a.x * b.x + a.y * b.y) * inv,
                       (a.y * b.x - a.x * b.y) * inv);
}

__global__ __launch_bounds__(64) void logdet_kernel(const float* __restrict__ n,
                                                    const float* __restrict__ Mf,
                                                    const float* __restrict__ yws,
                                                    float* __restrict__ out) {
    __shared__ float2 A[NSPIN][NF][NF + 1];   // +1 pad: dodge bank conflicts
    __shared__ float2 Fk[NSPIN][NF];
    __shared__ int    Ridx[NSPIN][NF];
    __shared__ float  res_abs[NSPIN];
    __shared__ float  res_ang[NSPIN];

    const int b    = blockIdx.x;
    const int s    = threadIdx.x >> 5;  // spin = wave id
    const int lane = threadIdx.x & 31;

    // Occupied-orbital index list (exactly NF ones per spin by construction).
    if (lane == 0) {
        int cnt = 0;
        const float* nr = n + (size_t)b * NV + s * N_ORB;
        for (int o = 0; o < N_ORB && cnt < NF; ++o)
            if (nr[o] > 0.5f) Ridx[s][cnt++] = o;
    }
    __syncthreads();

    // Gather A[i][j] = M[R[i]][j]  (complex64 interleaved)
    for (int i = 0; i < NF; ++i) {
        int r = Ridx[s][i];
        const float* mp = Mf + 2 * ((size_t)r * NF + lane);
        A[s][i][lane] = make_float2(mp[0], mp[1]);
    }
    __syncthreads();

    float labs = 0.f, ang = 0.f;
    int swaps = 0;
    for (int k = 0; k < NF; ++k) {
        // Partial pivot: argmax |A[i][k]|^2 over i >= k, wave-wide.
        float2 av = A[s][lane][k];
        float mag = (lane >= k) ? (av.x * av.x + av.y * av.y) : -1.0f;
        int idx = lane;
        for (int m = 16; m >= 1; m >>= 1) {
            float omag = __shfl_xor(mag, m, 32);
            int   oidx = __shfl_xor(idx, m, 32);
            if (omag > mag || (omag == mag && oidx < idx)) { mag = omag; idx = oidx; }
        }
        int p = idx;                       // uniform across the wave
        if (p != k) {                      // row swap (column-parallel)
            float2 t = A[s][k][lane];
            A[s][k][lane] = A[s][p][lane];
            A[s][p][lane] = t;
            swaps ^= 1;
        }
        __syncthreads();

        float2 piv = A[s][k][k];
        labs += 0.5f * logf(piv.x * piv.x + piv.y * piv.y);
        ang  += atan2f(piv.y, piv.x);

        if (lane > k) Fk[s][lane] = cdivf2(A[s][lane][k], piv);
        __syncthreads();

        if (lane > k) {                    // update trailing submatrix, col=lane
            float2 akj = A[s][k][lane];
            for (int i = k + 1; i < NF; ++i) {
                float2 f  = Fk[s][i];
                float2 ai = A[s][i][lane];
                ai.x -= f.x * akj.x - f.y * akj.y;
                ai.y -= f.x * akj.y + f.y * akj.x;
                A[s][i][lane] = ai;
            }
        }
        __syncthreads();
    }

    if (lane == 0) {
        // det phase: sum of pivot args + pi per swap; wrap to (-pi,pi] like slogdet
        float a = ang + 3.14159265358979f * (float)(swaps & 1);
        a = remainderf(a, 6.283185307179586f);
        res_abs[s] = labs;
        res_ang[s] = a;
    }
    __syncthreads();

    if (threadIdx.x == 0) {
        out[2 * b + 0] = yws[b] + res_abs[0] + res_abs[1];   // real
        out[2 * b + 1] = res_ang[0] + res_ang[1];            // imag
    }
}

// ---------------------------------------------------------------------------
extern "C" void kernel_launch(void* const* d_in, const int* in_sizes, int n_in,
                              void* d_out, int out_size, void* d_ws, size_t ws_size,
                              hipStream_t stream) {
    (void)in_sizes; (void)n_in; (void)out_size; (void)ws_size;

    const float* nptr = (const float*)d_in[0];   // (B, 128) float32
    const float* Wf   = (const float*)d_in[1];   // (128,128) complex64 interleaved
    const float* Mf   = (const float*)d_in[2];   // (64,32)   complex64 interleaved
    float* out = (float*)d_out;                  // B complex64 -> 2B floats

    float* G   = (float*)d_ws;                   // 128*128 f32 = 64 KB
    float* yws = G + NV * NV;                    // B f32 = 64 KB

    prep_g<<<(NV * NV + 255) / 256, 256, 0, stream>>>(Wf, G);
    quad_wmma<<<B_TOT / 16, 32, 0, stream>>>(nptr, G, yws);
    logdet_kernel<<<B_TOT, 64, 0, stream>>>(nptr, Mf, yws, out);
}